// KGTransformerPyG_27685359190570
// MI455X (gfx1250) — compile-verified
//
#include <hip/hip_runtime.h>
#include <math.h>

typedef float v2f __attribute__((ext_vector_type(2)));
typedef float v8f __attribute__((ext_vector_type(8)));

#define E_EDGES   4096
#define COMB_DIM  256
#define REL_DIM   128
#define IN_DIM    512      // COMB_DIM + 2*REL_DIM
#define HID       1024     // 2*IN_DIM
#define NUM_ENT   50000

// ---------------------------------------------------------------------------
// Gather + concat:  emb[e, :] = [combined_emb[head[e]], rel_embeds[t[e]],
//                                dynamic_relation_emb[t[e], 3, :, 1]]
// ---------------------------------------------------------------------------
__global__ __launch_bounds__(256) void build_emb_kernel(
    const float* __restrict__ comb, const float* __restrict__ rel,
    const float* __restrict__ dyn,  const int* __restrict__ edge_index,
    const int* __restrict__ edge_type, float* __restrict__ emb) {
  int idx = blockIdx.x * 256 + threadIdx.x;
  if (idx >= E_EDGES * IN_DIM) return;
  int e = idx >> 9;          // / IN_DIM (512)
  int d = idx & 511;
  float v;
  if (d < COMB_DIM) {
    int h = edge_index[e];                      // row 0 of (2, E)
    v = comb[(size_t)h * COMB_DIM + d];
  } else if (d < COMB_DIM + REL_DIM) {
    int r = edge_type[e];
    v = rel[(size_t)r * REL_DIM + (d - COMB_DIM)];
  } else {
    int r = edge_type[e];
    int dd = d - COMB_DIM - REL_DIM;
    // dynamic_relation_emb[r, 3, dd, 1], shape (200,4,128,2)
    v = dyn[(((size_t)r * 4 + 3) * REL_DIM + dd) * 2 + 1];
  }
  emb[idx] = v;
}

// ---------------------------------------------------------------------------
// FP32 WMMA GEMM:  C[M,N] = op(A[M,K] @ B[K,N] + bias), op = tanh or identity
// One wave -> 16 x (16*NTILES) strip of V_WMMA_F32_16X16X4_F32 tiles.
// 8 waves/block stacked in M -> 128 x (16*NTILES) macro tile.
// Branch-free (NTILES/KDIM/N are compile-time); remainder columns get a
// separate NTILES=1 launch, so EXEC is always full at every WMMA.
// Addressing: 3 per-lane 64-bit base pointers (A row, B rows khalf/khalf+1);
// tile offsets (64/128/192 B) and the 4-wide unrolled K advance
// (4*kk*N*4 B <= 2.4 MB < 2^23) live in the signed 24-bit load immediates,
// so the steady-state loop does only 3 pointer adds per 16 WMMAs.
// grid.x = M/128 varies fastest so all 32 M-tiles of one W2 N-panel run
// back-to-back and the panel is served from L2 (192 MB) after one HBM read.
// ---------------------------------------------------------------------------
template <int KDIM, int N, int NTILES, bool TANH>
__global__ __launch_bounds__(256) void wmma_gemm_kernel(
    const float* __restrict__ A, const float* __restrict__ B,
    const float* __restrict__ bias, float* __restrict__ C, int n_base) {
  const int lane  = threadIdx.x & 31;
  const int wave  = threadIdx.x >> 5;
  const int m0    = blockIdx.x * 128 + wave * 16;
  const int n0    = n_base + blockIdx.y * (16 * NTILES);
  const int lrow  = lane & 15;
  const int khalf = (lane >> 4) << 1;   // 0 for lanes 0-15, 2 for lanes 16-31

  v8f acc[NTILES];
#pragma unroll
  for (int t = 0; t < NTILES; ++t) acc[t] = (v8f){};

  // Per-lane base pointers; all further offsets are compile-time immediates.
  const float* ap  = A + (size_t)(m0 + lrow) * KDIM + khalf;  // A[m0+lrow, khalf]
  const float* bp0 = B + (size_t)khalf * N + (n0 + lrow);     // B[khalf,   col]
  const float* bp1 = bp0 + N;                                 // B[khalf+1, col]

  for (int k = 0; k < KDIM; k += 16) {   // 4 K-steps of 4 per iteration
#pragma unroll
    for (int kk = 0; kk < 4; ++kk) {
      v2f a = *(const v2f*)(ap + 4 * kk);           // imm offset 16*kk bytes
      v2f b[NTILES];
#pragma unroll
      for (int t = 0; t < NTILES; ++t) {
        b[t].x = bp0[(size_t)(4 * kk) * N + 16 * t]; // imm 4*kk*N*4 + 64*t
        b[t].y = bp1[(size_t)(4 * kk) * N + 16 * t];
      }
#pragma unroll
      for (int t = 0; t < NTILES; ++t)
        acc[t] = __builtin_amdgcn_wmma_f32_16x16x4_f32(
            false, a, false, b[t], (short)0, acc[t], false, false);
    }
    ap  += 16;
    bp0 += (size_t)16 * N;
    bp1 += (size_t)16 * N;
  }

  // C/D layout: VGPR r -> row m0 + r + 8*(lane>>4); column n0 + 16*t + lrow.
  const int rbase = m0 + ((lane >> 4) << 3);
  const int c0    = n0 + lrow;
  float* cp = C + (size_t)rbase * N + c0;   // store imms: r*N*4 + 64*t <= 1.4MB

#pragma unroll
  for (int t = 0; t < NTILES; ++t) {
    float bval = bias[c0 + 16 * t];
#pragma unroll
    for (int r = 0; r < 8; ++r) {
      float v = acc[t][r] + bval;
      if (TANH) v = tanhf(v);
      cp[(size_t)r * N + 16 * t] = v;
    }
  }
}

// ---------------------------------------------------------------------------
// Per-row online log-softmax stats + label gather (deterministic tree merge).
// rowlog[row] = pred[row, label] - max - log(sum exp(x - max))
// ---------------------------------------------------------------------------
__global__ __launch_bounds__(256) void row_logsoftmax_kernel(
    const float* __restrict__ pred, const int* __restrict__ target_tails,
    const int* __restrict__ node_id, float* __restrict__ rowlog) {
  __shared__ float sm[256];
  __shared__ float ss[256];
  const int row = blockIdx.x;
  const int tid = threadIdx.x;
  const float* p = pred + (size_t)row * NUM_ENT;

  float m = -INFINITY, s = 0.0f;
  for (int i = tid; i < NUM_ENT; i += 256) {
    float x  = p[i];
    float nm = fmaxf(m, x);
    s = s * __expf(m - nm) + __expf(x - nm);
    m = nm;
  }
  sm[tid] = m; ss[tid] = s;
  __syncthreads();
  for (int off = 128; off > 0; off >>= 1) {
    if (tid < off) {
      float m2 = sm[tid + off], s2 = ss[tid + off];
      float nm = fmaxf(sm[tid], m2);
      ss[tid]  = ss[tid] * __expf(sm[tid] - nm) + s2 * __expf(m2 - nm);
      sm[tid]  = nm;
    }
    __syncthreads();
  }
  if (tid == 0) {
    int label   = node_id[target_tails[row]];
    rowlog[row] = p[label] - sm[0] - logf(ss[0]);
  }
}

__global__ __launch_bounds__(256) void mean_kernel(
    const float* __restrict__ rowlog, float* __restrict__ out) {
  __shared__ float sh[256];
  float s = 0.0f;
  for (int i = threadIdx.x; i < E_EDGES; i += 256) s += rowlog[i];
  sh[threadIdx.x] = s;
  __syncthreads();
  for (int off = 128; off > 0; off >>= 1) {
    if (threadIdx.x < off) sh[threadIdx.x] += sh[threadIdx.x + off];
    __syncthreads();
  }
  if (threadIdx.x == 0) out[0] = sh[0] / (float)E_EDGES;
}

// ---------------------------------------------------------------------------
extern "C" void kernel_launch(void* const* d_in, const int* in_sizes, int n_in,
                              void* d_out, int out_size, void* d_ws, size_t ws_size,
                              hipStream_t stream) {
  const float* comb = (const float*)d_in[0];   // (20000, 256)
  const float* rel  = (const float*)d_in[1];   // (200, 128)
  const float* dyn  = (const float*)d_in[2];   // (200, 4, 128, 2)
  const float* W1   = (const float*)d_in[3];   // (512, 1024)
  const float* b1   = (const float*)d_in[4];   // (1024,)
  const float* W2   = (const float*)d_in[5];   // (1024, 50000)
  const float* b2   = (const float*)d_in[6];   // (50000,)
  const int* edge_index   = (const int*)d_in[7];  // (2, 4096)
  const int* edge_type    = (const int*)d_in[8];  // (4096,)
  const int* target_tails = (const int*)d_in[9];  // (4096,)
  const int* node_id      = (const int*)d_in[10]; // (20000,)

  float* out       = (float*)d_out;
  float* tail_pred = out + 1;                  // (4096, 50000) row-major

  float* emb    = (float*)d_ws;                          // 4096*512
  float* h      = emb + (size_t)E_EDGES * IN_DIM;        // 4096*1024
  float* rowlog = h + (size_t)E_EDGES * HID;             // 4096

  build_emb_kernel<<<(E_EDGES * IN_DIM + 255) / 256, 256, 0, stream>>>(
      comb, rel, dyn, edge_index, edge_type, emb);

  // GEMM1: 4096x512 @ 512x1024, tanh epilogue. N divisible by 64.
  dim3 g1(E_EDGES / 128, HID / 64);
  wmma_gemm_kernel<IN_DIM, HID, 4, true>
      <<<g1, 256, 0, stream>>>(emb, W1, b1, h, 0);

  // GEMM2 main: columns [0, 49984) in 64-wide strips, branch-free.
  dim3 g2(E_EDGES / 128, NUM_ENT / 64);   // 781 full strips
  wmma_gemm_kernel<HID, NUM_ENT, 4, false>
      <<<g2, 256, 0, stream>>>(h, W2, b2, tail_pred, 0);

  // GEMM2 remainder: columns [49984, 50000) -> one 16-wide strip.
  dim3 g2r(E_EDGES / 128, 1);
  wmma_gemm_kernel<HID, NUM_ENT, 1, false>
      <<<g2r, 256, 0, stream>>>(h, W2, b2, tail_pred, (NUM_ENT / 64) * 64);

  row_logsoftmax_kernel<<<E_EDGES, 256, 0, stream>>>(tail_pred, target_tails,
                                                     node_id, rowlog);
  mean_kernel<<<1, 256, 0, stream>>>(rowlog, out);
}